// SeqBatteryEnv_26104811225476
// MI455X (gfx1250) — compile-verified
//
#include <hip/hip_runtime.h>
#include <stdint.h>

// Battery rollout: B=4096 independent sequential scans over T=4096 steps.
// 1 lane per batch row, 128 single-wave workgroups, async global->LDS
// double-buffered tile pipeline (CDNA5 ASYNCcnt path), GVS addressing.
// Recurrence algebraically restructured to a 5-op dependent chain:
//   ns = clamp( min( fma(K, agr', after), s + 5K ), 0, CAP )
// which is exactly equivalent (in reals) to the reference clip sequence.

#define B_ROWS 4096
#define T_STEPS 4096
#define TS 32                 // time-chunk (steps per tile)
#define NC (T_STEPS / TS)     // 128 chunks
#define PAD 36                // padded LDS row stride in floats: 144B (16B-aligned, 2-way conflicts max)

typedef float v4f __attribute__((ext_vector_type(4)));

__global__ __launch_bounds__(32) void battery_scan_kernel(
    const float* __restrict__ ga,   // grid_action  [B,T]
    const float* __restrict__ pa,   // pv_action    [B,T]
    const float* __restrict__ pp,   // pv_power     [B,T]
    const float* __restrict__ pr,   // price        [B,T]
    float* __restrict__ out)        // trace [B,T+1] ++ cost [B,T]
{
  // 2 buffers x 4 inputs x 32 rows x PAD floats = 36864 bytes
  __shared__ float smem[2][4][32 * PAD];

  const int lane = threadIdx.x;          // 0..31, one batch row per lane
  const int r0   = blockIdx.x * 32;
  const int row  = r0 + lane;

  // Coalesced async-load mapping: lane l fetches 16B of row (i*4 + l/8) at
  // float offset (l&7)*4 -> each b128 instruction covers 4 consecutive rows'
  // contiguous 128B chunk segments, fully coalesced.
  const int rl_sub = lane >> 3;
  const int toff   = (lane & 7) * 4;

  const float* bases[4] = { ga, pa, pp, pr };

  auto issue_chunk = [&](int buf, int c) {
    #pragma unroll
    for (int inp = 0; inp < 4; ++inp) {
      // uniform 64-bit base: only c-dependent part of the address (SALU only)
      unsigned long long sb = (unsigned long long)(uintptr_t)
          (bases[inp] + (((size_t)r0) << 12) + (size_t)(c * TS));
      #pragma unroll
      for (int i = 0; i < 8; ++i) {
        const int rl = i * 4 + rl_sub;
        // chunk-invariant per-lane 32-bit byte offset (hoisted by LICM)
        unsigned vo = ((unsigned)rl << 14) + (unsigned)(toff * 4);
        // chunk-invariant LDS destination byte address (hoisted by LICM)
        unsigned ld = (unsigned)(uintptr_t)&smem[buf][inp][rl * PAD + toff];
        // CDNA5 async DMA, GVS mode: mem = SGPR base + 32-bit VGPR offset.
        // Inputs are single-use streams -> non-temporal load hint.
        asm volatile("global_load_async_to_lds_b128 %0, %1, %2 th:TH_LOAD_NT"
                     :: "v"(ld), "v"(vo), "s"(sb) : "memory");
      }
    }
  };

  const float CAP   = 13.5f;
  const float K     = (float)(5.0 / 60.0);   // kW -> kWh per 5-min step
  const float FIVEK = 5.0f * (float)(5.0 / 60.0);
  const float PS    = (float)(1.0 / 1000.0);

  float s = 6.75f;                                  // INITIAL_CHARGE_KWH
  float* trp = out + (size_t)row * (T_STEPS + 1);   // trace row
  float* cop = out + (size_t)B_ROWS * (T_STEPS + 1)
                   + (size_t)row * T_STEPS;         // cost row
  __builtin_nontemporal_store(6.75f, trp);          // trace[:,0] = init
  trp += 1;

  auto process = [&](int buf, int c) {
    const float* Gm = &smem[buf][0][lane * PAD];
    const float* Am = &smem[buf][1][lane * PAD];
    const float* Pm = &smem[buf][2][lane * PAD];
    const float* Rm = &smem[buf][3][lane * PAD];
    float* t_out = trp + (size_t)c * TS;
    float* c_out = cop + (size_t)c * TS;
    #pragma unroll
    for (int q = 0; q < TS; q += 4) {
      v4f g4 = *(const v4f*)(Gm + q);   // ds_load_b128, 16B-aligned
      v4f a4 = *(const v4f*)(Am + q);
      v4f p4 = *(const v4f*)(Pm + q);
      v4f r4 = *(const v4f*)(Rm + q);
      v4f c4 = {};
      #pragma unroll
      for (int u = 0; u < 4; ++u) {
        // ---- state-independent (dual-issues alongside the chain) ----
        float apv = a4[u] * p4[u];
        float ppc = fminf(fmaxf(apv, 0.0f), 5.0f) * K;   // provisional pv kWh (>=0)
        float agr = fmaxf(g4[u] * 5.0f, -5.0f);          // grid rate, low-clipped
        float pvk = p4[u] * K;
        // ---- sequential recurrence: 5-op dependent chain ----
        float slim  = s + FIVEK;                  // parallel branch off s
        float uu    = s + ppc;                    // dep 1
        float after = fminf(uu, CAP);             // dep 2 (uu >= 0 always)
        float cand  = fmaf(K, agr, after);        // dep 3
        float pre   = fminf(cand, slim);          // dep 4  == after + K*realised_grid
        float ns    = fminf(fmaxf(pre, 0.0f), CAP); // dep 5 (v_med3)
        // ---- side outputs (off the chain) ----
        float amt   = after - s;                  // realised pv kWh
        float gamt  = ns - after;                 // grid kWh
        float pvex  = amt - pvk;                  // pv export kWh
        c4[u] = (r4[u] * PS) * (gamt + pvex);
        s = ns;
        // trace row stride is 4097 floats -> 4B-aligned scalar NT store
        __builtin_nontemporal_store(ns, t_out + q + u);
      }
      // cost row stride 4096 floats -> 16B-aligned vector NT store
      __builtin_nontemporal_store(c4, (v4f*)(c_out + q));
    }
  };

  // software pipeline: prefetch chunk c+1 while scanning chunk c
  issue_chunk(0, 0);
  for (int c = 0; c < NC - 1; ++c) {
    issue_chunk((c + 1) & 1, c + 1);
    // only the 32 loads of chunk c+1 may remain outstanding => chunk c in LDS
    asm volatile("s_wait_asynccnt 32" ::: "memory");
    process(c & 1, c);
  }
  asm volatile("s_wait_asynccnt 0" ::: "memory");
  process((NC - 1) & 1, NC - 1);
}

extern "C" void kernel_launch(void* const* d_in, const int* in_sizes, int n_in,
                              void* d_out, int out_size, void* d_ws, size_t ws_size,
                              hipStream_t stream)
{
  (void)in_sizes; (void)n_in; (void)d_ws; (void)ws_size; (void)out_size;
  const float* ga = (const float*)d_in[0];  // grid_action
  const float* pa = (const float*)d_in[1];  // pv_action
  const float* pp = (const float*)d_in[2];  // pv_power
  const float* pr = (const float*)d_in[3];  // price
  float* out = (float*)d_out;               // trace [B,T+1] then cost [B,T]
  battery_scan_kernel<<<B_ROWS / 32, 32, 0, stream>>>(ga, pa, pp, pr, out);
}